// SlotAttention_64398739636839
// MI455X (gfx1250) — compile-verified
//
#include <hip/hip_runtime.h>
#include <hip/hip_bf16.h>

typedef float v2f __attribute__((ext_vector_type(2)));
typedef float v4f __attribute__((ext_vector_type(4)));
typedef float v8f __attribute__((ext_vector_type(8)));

constexpr int kB = 32;
constexpr int kN = 4096;
constexpr int kD = 256;
constexpr int kK = 8;
constexpr int kS = 16;
constexpr int kH = 128;
constexpr int kIter = 3;
#define EPS_F 1e-8f
#define LN_EPS_F 1e-5f

// ---------------------------------------------------------------------------
// Kernel 1: fused LayerNorm(inputs) + k/v projection via V_WMMA_F32_16X16X4_F32
//
// One wave per 16-token tile. Column remap for b128 loads: lanes 0-15 supply
// columns [g, g+4) as K=0,1 over two WMMA steps; lanes 16-31 supply columns
// [g+4, g+8) as K=2,3. Each operand (x, ln_w, ln_b, wk-row, wv-row) is one
// global_load_b128 serving two WMMA steps. WMMA K-order is associative-free,
// so the result matches the reference GEMM up to fp summation order.
// LN stats: lane covers cols ≡ [off, off+4) mod 8; partner lane (L^16) covers
// the rest; one __shfl_xor(...,16) combines. No LDS, EXEC all-ones.
// ---------------------------------------------------------------------------
__global__ __launch_bounds__(64) void kv_proj_kernel(
    const float* __restrict__ inp,
    const float* __restrict__ lnw, const float* __restrict__ lnb,
    const float* __restrict__ wk,  const float* __restrict__ wv,
    float* __restrict__ kout, float* __restrict__ vout)
{
    const int lane = threadIdx.x & 31;
    const int wave = threadIdx.x >> 5;
    const int tile = blockIdx.x * 2 + wave;          // 4096 blocks * 2 waves = 8192 tiles
    const int arow = lane & 15;                      // A-matrix row within tile (token)
    const int off  = (lane >> 4) * 4;                // column offset for this half-wave
    const float* xrow = inp + (size_t)(tile * 16 + arow) * kD;

    // pass 1: row mean / variance (this lane's 128 columns, then combine halves)
    float sum = 0.f, sq = 0.f;
    #pragma unroll 8
    for (int c = off; c < kD; c += 8) {
        v4f x = *(const v4f*)(xrow + c);
        sum += (x.x + x.y) + (x.z + x.w);
        sq  += (x.x * x.x + x.y * x.y) + (x.z * x.z + x.w * x.w);
    }
    sum += __shfl_xor(sum, 16);
    sq  += __shfl_xor(sq,  16);
    const float mu   = sum * (1.0f / kD);
    const float rstd = rsqrtf(sq * (1.0f / kD) - mu * mu + LN_EPS_F);

    // B-matrix: column n = lane&15 is weight row n; same col offsets as A.
    const float* wkrow = wk + arow * kD;
    const float* wvrow = wv + arow * kD;

    v8f ck = {0.f,0.f,0.f,0.f,0.f,0.f,0.f,0.f};
    v8f cv = {0.f,0.f,0.f,0.f,0.f,0.f,0.f,0.f};

    #pragma unroll 4
    for (int g = 0; g < kD; g += 8) {
        const int c = g + off;
        v4f x  = *(const v4f*)(xrow  + c);
        v4f w  = *(const v4f*)(lnw   + c);
        v4f bb = *(const v4f*)(lnb   + c);
        v4f bk = *(const v4f*)(wkrow + c);
        v4f bv = *(const v4f*)(wvrow + c);
        v2f a0, a1;
        a0.x = (x.x - mu) * rstd * w.x + bb.x;   // col c+0  (K=0 low / K=2 high)
        a0.y = (x.y - mu) * rstd * w.y + bb.y;   // col c+1  (K=1 low / K=3 high)
        a1.x = (x.z - mu) * rstd * w.z + bb.z;   // col c+2
        a1.y = (x.w - mu) * rstd * w.w + bb.w;   // col c+3
        v2f bk0 = {bk.x, bk.y}, bk1 = {bk.z, bk.w};
        v2f bv0 = {bv.x, bv.y}, bv1 = {bv.z, bv.w};
        ck = __builtin_amdgcn_wmma_f32_16x16x4_f32(false, a0, false, bk0, (short)0, ck, false, false);
        cv = __builtin_amdgcn_wmma_f32_16x16x4_f32(false, a0, false, bv0, (short)0, cv, false, false);
        ck = __builtin_amdgcn_wmma_f32_16x16x4_f32(false, a1, false, bk1, (short)0, ck, false, false);
        cv = __builtin_amdgcn_wmma_f32_16x16x4_f32(false, a1, false, bv1, (short)0, cv, false, false);
    }

    // D layout: lanes 0-15 -> M=j, lanes 16-31 -> M=j+8; N = lane&15.
    const int n     = arow;
    const int mbase = tile * 16 + (lane >> 4) * 8;
    #pragma unroll
    for (int j = 0; j < 8; ++j) {
        kout[(size_t)(mbase + j) * kS + n] = ck[j];
        vout[(size_t)(mbase + j) * kS + n] = cv[j];
    }
}

__device__ __forceinline__ void ld_row16(const float* __restrict__ p, float* r) {
    const float4* q = (const float4*)p;
    float4 a = q[0], b = q[1], c = q[2], d = q[3];
    r[0]=a.x; r[1]=a.y; r[2]=a.z; r[3]=a.w;
    r[4]=b.x; r[5]=b.y; r[6]=b.z; r[7]=b.w;
    r[8]=c.x; r[9]=c.y; r[10]=c.z; r[11]=c.w;
    r[12]=d.x; r[13]=d.y; r[14]=d.z; r[15]=d.w;
}

// ---------------------------------------------------------------------------
// Kernel 2: 3 slot-attention iterations. One block per batch, 512 threads.
// ---------------------------------------------------------------------------
__global__ __launch_bounds__(512) void slot_iter_kernel(
    const float* __restrict__ init_slots,
    const float* __restrict__ ln_s_w, const float* __restrict__ ln_s_b,
    const float* __restrict__ ln_m_w, const float* __restrict__ ln_m_b,
    const float* __restrict__ wq,
    const float* __restrict__ w_ih, const float* __restrict__ w_hh,
    const float* __restrict__ b_ih, const float* __restrict__ b_hh,
    const float* __restrict__ w1, const float* __restrict__ b1,
    const float* __restrict__ w2, const float* __restrict__ b2,
    const float* __restrict__ kbuf, const float* __restrict__ vbuf,
    float* __restrict__ slots_out, float* __restrict__ attn_out)
{
    __shared__ float slots_sh[kK][kS];
    __shared__ float y_sh[kK][kS];
    __shared__ float q_sh[kK][kS];
    __shared__ float U_sh[kK][kS];
    __shared__ float S1_sh[kK];
    __shared__ float upd_sh[kK][kS];
    __shared__ float h_sh[kK][kS];
    __shared__ float hid_sh[kK][kH];

    const int b = blockIdx.x;
    const int t = threadIdx.x;
    const int lane = t & 31;
    const float* kb = kbuf + (size_t)b * kN * kS;
    const float* vb = vbuf + (size_t)b * kN * kS;
    float* attn_b = attn_out + (size_t)b * kK * kN;

    if (t < kK * kS) slots_sh[t >> 4][t & 15] = init_slots[b * kK * kS + t];
    __syncthreads();

    for (int it = 0; it < kIter; ++it) {
        const bool last = (it == kIter - 1);
        // zero reduction buffers + LayerNorm(slots) -> y_sh
        if (t < kK * kS) U_sh[t >> 4][t & 15] = 0.f;
        if (t >= 128 && t < 128 + kK) S1_sh[t - 128] = 0.f;
        if (t >= 160 && t < 160 + kK) {
            const int kk = t - 160;
            float mu = 0.f;
            #pragma unroll
            for (int s = 0; s < kS; ++s) mu += slots_sh[kk][s];
            mu *= (1.f / kS);
            float var = 0.f;
            #pragma unroll
            for (int s = 0; s < kS; ++s) { float d = slots_sh[kk][s] - mu; var += d * d; }
            const float r = rsqrtf(var * (1.f / kS) + LN_EPS_F);
            #pragma unroll
            for (int s = 0; s < kS; ++s)
                y_sh[kk][s] = (slots_sh[kk][s] - mu) * r * ln_s_w[s] + ln_s_b[s];
        }
        __syncthreads();
        // q = LN(slots) @ wq^T
        if (t < kK * kS) {
            const int kk = t >> 4, tt = t & 15;
            float acc = 0.f;
            #pragma unroll
            for (int s = 0; s < kS; ++s) acc += y_sh[kk][s] * wq[tt * kS + s];
            q_sh[kk][tt] = acc;
        }
        __syncthreads();

        // attention pass over N: softmax over slots, accumulate sums + updates
        float Ul[kK][kS];
        float S1l[kK];
        #pragma unroll
        for (int kk = 0; kk < kK; ++kk) {
            S1l[kk] = 0.f;
            #pragma unroll
            for (int s = 0; s < kS; ++s) Ul[kk][s] = 0.f;
        }
        for (int n = t; n < kN; n += 512) {
            float kv_[kS], vv_[kS];
            ld_row16(kb + (size_t)n * kS, kv_);
            ld_row16(vb + (size_t)n * kS, vv_);
            float lg[kK];
            #pragma unroll
            for (int kk = 0; kk < kK; ++kk) {
                float a = 0.f;
                #pragma unroll
                for (int s = 0; s < kS; ++s) a += q_sh[kk][s] * kv_[s];
                lg[kk] = a * 0.25f;  // SLOT_DIM^-0.5
            }
            float m = lg[0];
            #pragma unroll
            for (int kk = 1; kk < kK; ++kk) m = fmaxf(m, lg[kk]);
            float p[kK]; float es = 0.f;
            #pragma unroll
            for (int kk = 0; kk < kK; ++kk) { p[kk] = __expf(lg[kk] - m); es += p[kk]; }
            const float inv = 1.f / es;
            #pragma unroll
            for (int kk = 0; kk < kK; ++kk) {
                p[kk] = p[kk] * inv + EPS_F;
                S1l[kk] += p[kk];
                #pragma unroll
                for (int s = 0; s < kS; ++s) Ul[kk][s] += p[kk] * vv_[s];
                if (last) attn_b[kk * kN + n] = p[kk];
            }
        }
        // wave reduction then LDS atomics
        #pragma unroll
        for (int kk = 0; kk < kK; ++kk) {
            #pragma unroll
            for (int w = 16; w >= 1; w >>= 1) S1l[kk] += __shfl_xor(S1l[kk], w);
            #pragma unroll
            for (int s = 0; s < kS; ++s) {
                #pragma unroll
                for (int w = 16; w >= 1; w >>= 1) Ul[kk][s] += __shfl_xor(Ul[kk][s], w);
            }
        }
        if (lane == 0) {
            #pragma unroll
            for (int kk = 0; kk < kK; ++kk) {
                atomicAdd(&S1_sh[kk], S1l[kk]);
                #pragma unroll
                for (int s = 0; s < kS; ++s) atomicAdd(&U_sh[kk][s], Ul[kk][s]);
            }
        }
        __syncthreads();

        // final-iteration attn rescale by 1/sum_n
        if (last) {
            for (int idx = t; idx < kK * kN; idx += 512) {
                const int kk = idx >> 12;  // /4096
                attn_b[idx] *= 1.f / S1_sh[kk];
            }
        }
        // updates = U / S1
        if (t < kK * kS) {
            const int kk = t >> 4, s = t & 15;
            upd_sh[kk][s] = U_sh[kk][s] / S1_sh[kk];
        }
        __syncthreads();
        // GRU cell (gates ordered r,z,n)
        if (t < kK * kS) {
            const int kk = t >> 4, s = t & 15;
            float gi_r = b_ih[s], gi_z = b_ih[kS + s], gi_n = b_ih[2 * kS + s];
            float gh_r = b_hh[s], gh_z = b_hh[kS + s], gh_n = b_hh[2 * kS + s];
            #pragma unroll
            for (int j = 0; j < kS; ++j) {
                const float x = upd_sh[kk][j];
                const float h = slots_sh[kk][j];
                gi_r += x * w_ih[s * kS + j];
                gi_z += x * w_ih[(kS + s) * kS + j];
                gi_n += x * w_ih[(2 * kS + s) * kS + j];
                gh_r += h * w_hh[s * kS + j];
                gh_z += h * w_hh[(kS + s) * kS + j];
                gh_n += h * w_hh[(2 * kS + s) * kS + j];
            }
            const float r = 1.f / (1.f + __expf(-(gi_r + gh_r)));
            const float z = 1.f / (1.f + __expf(-(gi_z + gh_z)));
            const float nn = tanhf(gi_n + r * gh_n);
            h_sh[kk][s] = (1.f - z) * nn + z * slots_sh[kk][s];
        }
        __syncthreads();
        // LayerNorm(h) with ln_m -> y_sh
        if (t < kK) {
            const int kk = t;
            float mu = 0.f;
            #pragma unroll
            for (int s = 0; s < kS; ++s) mu += h_sh[kk][s];
            mu *= (1.f / kS);
            float var = 0.f;
            #pragma unroll
            for (int s = 0; s < kS; ++s) { float d = h_sh[kk][s] - mu; var += d * d; }
            const float r = rsqrtf(var * (1.f / kS) + LN_EPS_F);
            #pragma unroll
            for (int s = 0; s < kS; ++s)
                y_sh[kk][s] = (h_sh[kk][s] - mu) * r * ln_m_w[s] + ln_m_b[s];
        }
        __syncthreads();
        // MLP hidden: relu(y @ w1^T + b1)
        for (int idx = t; idx < kK * kH; idx += 512) {
            const int kk = idx >> 7, hh = idx & 127;
            float a = b1[hh];
            #pragma unroll
            for (int s = 0; s < kS; ++s) a += y_sh[kk][s] * w1[hh * kS + s];
            hid_sh[kk][hh] = fmaxf(a, 0.f);
        }
        __syncthreads();
        // MLP out + residual -> new slots
        if (t < kK * kS) {
            const int kk = t >> 4, s = t & 15;
            float a = b2[s];
            #pragma unroll 8
            for (int hh = 0; hh < kH; ++hh) a += hid_sh[kk][hh] * w2[s * kH + hh];
            slots_sh[kk][s] = h_sh[kk][s] + a;
        }
        __syncthreads();
    }

    if (t < kK * kS) slots_out[b * kK * kS + t] = slots_sh[t >> 4][t & 15];
}

extern "C" void kernel_launch(void* const* d_in, const int* in_sizes, int n_in,
                              void* d_out, int out_size, void* d_ws, size_t ws_size,
                              hipStream_t stream) {
    (void)in_sizes; (void)n_in; (void)out_size; (void)ws_size;
    const float* inputs     = (const float*)d_in[0];
    const float* init_slots = (const float*)d_in[1];
    const float* ln_in_w    = (const float*)d_in[2];
    const float* ln_in_b    = (const float*)d_in[3];
    const float* ln_s_w     = (const float*)d_in[4];
    const float* ln_s_b     = (const float*)d_in[5];
    const float* ln_m_w     = (const float*)d_in[6];
    const float* ln_m_b     = (const float*)d_in[7];
    const float* wq         = (const float*)d_in[8];
    const float* wk         = (const float*)d_in[9];
    const float* wv         = (const float*)d_in[10];
    const float* w_ih       = (const float*)d_in[11];
    const float* w_hh       = (const float*)d_in[12];
    const float* b_ih       = (const float*)d_in[13];
    const float* b_hh       = (const float*)d_in[14];
    const float* mlp_w1     = (const float*)d_in[15];
    const float* mlp_b1     = (const float*)d_in[16];
    const float* mlp_w2     = (const float*)d_in[17];
    const float* mlp_b2     = (const float*)d_in[18];

    float* kbuf = (float*)d_ws;                       // [B,N,S] = 2M floats
    float* vbuf = kbuf + (size_t)kB * kN * kS;        // [B,N,S]
    float* slots_out = (float*)d_out;                 // [B,K,S] = 4096 floats
    float* attn_out  = slots_out + kB * kK * kS;      // [B,K,N] = 1M floats

    // 8192 row-tiles of 16 tokens, 2 waves (tiles) per 64-thread block
    kv_proj_kernel<<<(kB * kN) / 32, 64, 0, stream>>>(
        inputs, ln_in_w, ln_in_b, wk, wv, kbuf, vbuf);

    slot_iter_kernel<<<kB, 512, 0, stream>>>(
        init_slots, ln_s_w, ln_s_b, ln_m_w, ln_m_b, wq,
        w_ih, w_hh, b_ih, b_hh, mlp_w1, mlp_b1, mlp_w2, mlp_b2,
        kbuf, vbuf, slots_out, attn_out);
}